// RandomSample_33612414058918
// MI455X (gfx1250) — compile-verified
//
#include <hip/hip_runtime.h>

typedef __attribute__((ext_vector_type(2))) float v2f;
typedef __attribute__((ext_vector_type(8))) float v8f;

#define BATCH 4
#define NTOT  16384
#define HALF  8192
#define FDIM  32

#define OUT_PC_OFF   0
#define OUT_PC_N     (BATCH*HALF*3)            /* 98304   */
#define OUT_FE_OFF   (OUT_PC_OFF + OUT_PC_N)   /* 98304   */
#define OUT_FE_N     (BATCH*HALF*FDIM)         /* 1048576 */
#define OUT_IDX_OFF  (OUT_FE_OFF + OUT_FE_N)   /* 1146880 */
#define OUT_IDX_N    (BATCH*HALF)              /* 32768   */
#define OUT_RND_OFF  (OUT_IDX_OFF + OUT_IDX_N) /* 1179648 */
#define OUT_RND_N    (NTOT)                    /* 16384   */

#define GATHER_THREADS (OUT_PC_N + OUT_FE_N + OUT_RND_N) /* 1163264 */

// ---------------------------------------------------------------------------
// Kernel 1: pure gather. Writes valid_pc, valid_feats, rnds-as-float to d_out.
// ---------------------------------------------------------------------------
__global__ __launch_bounds__(256) void gather_kernel(
    const float* __restrict__ pc, const float* __restrict__ feats,
    const int* __restrict__ rnds, float* __restrict__ out) {
  int t = blockIdx.x * 256 + threadIdx.x;
  if (t < OUT_PC_N) {
    int k = t % 3;
    int i = (t / 3) % HALF;
    int b = t / (3 * HALF);
    int vi = rnds[i];
    out[OUT_PC_OFF + t] = pc[(b * NTOT + vi) * 3 + k];
  } else if (t < OUT_PC_N + OUT_FE_N) {
    int o = t - OUT_PC_N;
    int f = o % FDIM;
    int i = (o / FDIM) % HALF;
    int b = o / (FDIM * HALF);
    int vi = rnds[i];
    out[OUT_FE_OFF + o] = feats[(b * NTOT + vi) * FDIM + f];
  } else if (t < GATHER_THREADS) {
    int o = t - (OUT_PC_N + OUT_FE_N);
    out[OUT_RND_OFF + o] = (float)rnds[o];
  }
}

// ---------------------------------------------------------------------------
// Kernel 2: KNN via V_WMMA_F32_16X16X4_F32.
// Score(q,p) = |p|^2 - 2 q.p  (|q|^2 dropped: constant per row, preserves
// ordering).  A row = (-2qx,-2qy,-2qz,1), B col = (px,py,pz,|p|^2), K=4.
// One block = 8 waves = 128 queries of one batch; points staged in LDS.
// ---------------------------------------------------------------------------
#define CHUNK 4096  /* points per LDS stage: 4096 * 16B = 64KB */

// Branchless pair compare: bitwise | / & so clang emits v_cmp + mask ops,
// no EXEC-mask control flow in the hot loop.
__device__ __forceinline__ bool pair_less(float v, int i, float w, int j) {
  return (bool)((v < w) | ((v == w) & (i < j)));
}

__global__ __launch_bounds__(256) void knn_kernel(
    const float* __restrict__ pc, const int* __restrict__ rnds,
    float* __restrict__ out) {
  __shared__ float4 spts[CHUNK];  // (px,py,pz,|p|^2)

  const int tid  = threadIdx.x;
  const int wave = tid >> 5;
  const int lane = tid & 31;
  const int lo   = lane & 15;
  const bool hi  = lane >= 16;

  const int b  = blockIdx.x >> 6;                         // 64 blocks per batch
  const int q0 = ((blockIdx.x & 63) << 7) + (wave << 4);  // 16 queries/wave

  // ---- A operand: 16 queries, augmented, ISA 16x4 f32 layout ----
  // VGPR0: K=0 (lanes 0-15) / K=2 (lanes 16-31); VGPR1: K=1 / K=3.
  int qi = rnds[HALF + q0 + lo];
  const float* qp = pc + (size_t)(b * NTOT + qi) * 3;
  v2f Amat;
  if (!hi) { Amat.x = -2.0f * qp[0]; Amat.y = -2.0f * qp[1]; }
  else     { Amat.x = -2.0f * qp[2]; Amat.y = 1.0f; }

  const float INF = 3.402823466e38f;
  float min1[8], min2[8];
  int   idx1[8], idx2[8];
#pragma unroll
  for (int e = 0; e < 8; ++e) { min1[e] = INF; min2[e] = INF; idx1[e] = 0; idx2[e] = 0; }

  // LDS float offset within a point for this lane's B fragment.
  const int boff = hi ? 2 : 0;

  for (int chunk = 0; chunk < HALF / CHUNK; ++chunk) {
    __syncthreads();
    // Stage CHUNK augmented valid points into LDS (gather via rnds).
    for (int i = tid; i < CHUNK; i += 256) {
      int vi = rnds[chunk * CHUNK + i];
      const float* pp = pc + (size_t)(b * NTOT + vi) * 3;
      float x = pp[0], y = pp[1], z = pp[2];
      spts[i] = make_float4(x, y, z, x * x + y * y + z * z);
    }
    __syncthreads();

    // Software-pipelined B-tile load: fetch tile t0+16 while updating tile t0.
    // B operand: 4x16 f32, mirrored A layout: VGPR0 K=0/K=2, VGPR1 K=1/K=3.
    v2f Bnext;
    {
      const float* sp = reinterpret_cast<const float*>(&spts[lo]) + boff;
      Bnext.x = sp[0];
      Bnext.y = sp[1];
    }

    for (int t0 = 0; t0 < CHUNK; t0 += 16) {
      v2f Bcur = Bnext;
      {
        int tn = (t0 + 16) & (CHUNK - 1);  // wrap: always in-range, no branch
        const float* sp = reinterpret_cast<const float*>(&spts[tn + lo]) + boff;
        Bnext.x = sp[0];
        Bnext.y = sp[1];
      }

      v8f c = {};
      c = __builtin_amdgcn_wmma_f32_16x16x4_f32(
          /*neg_a=*/false, Amat, /*neg_b=*/false, Bcur,
          /*c_mod=*/(short)0, c, /*reuse_a=*/false, /*reuse_b=*/false);

      const int col = chunk * CHUNK + t0 + lo;
#pragma unroll
      for (int e = 0; e < 8; ++e) {
        float v = c[e];
        bool lt1    = v < min1[e];
        float disp  = lt1 ? min1[e] : v;
        int   dispI = lt1 ? idx1[e] : col;
        min1[e]     = lt1 ? v : min1[e];
        idx1[e]     = lt1 ? col : idx1[e];
        bool lt2    = pair_less(disp, dispI, min2[e], idx2[e]);
        min2[e]     = lt2 ? disp : min2[e];
        idx2[e]     = lt2 ? dispI : idx2[e];
      }
    }
  }

  // ---- Cross-lane top-2 merge within each 16-lane half (xor butterfly) ----
#pragma unroll
  for (int e = 0; e < 8; ++e) {
    float m1 = min1[e], m2 = min2[e];
    int   i1 = idx1[e], i2 = idx2[e];
#pragma unroll
    for (int m = 8; m >= 1; m >>= 1) {
      float om1 = __shfl_xor(m1, m, 32);
      float om2 = __shfl_xor(m2, m, 32);
      int   oi1 = __shfl_xor(i1, m, 32);
      int   oi2 = __shfl_xor(i2, m, 32);
      // Merge two sorted pairs -> top-2 of the union (stable on index).
      bool aFirst  = pair_less(m1, i1, om1, oi1);
      float n1     = aFirst ? m1 : om1;
      int   ni1    = aFirst ? i1 : oi1;
      float candA  = aFirst ? om1 : m1;   // loser of the firsts
      int   candAi = aFirst ? oi1 : i1;
      float candB  = aFirst ? m2 : om2;   // winner's second
      int   candBi = aFirst ? i2 : oi2;
      bool bFirst  = pair_less(candA, candAi, candB, candBi);
      m2 = bFirst ? candA : candB;
      i2 = bFirst ? candAi : candBi;
      m1 = n1; i1 = ni1;
    }
    if (lo == 0) {  // lane 0 -> row e, lane 16 -> row e+8
      int row = e + (hi ? 8 : 0);
      int q   = q0 + row;
      out[OUT_IDX_OFF + b * HALF + q] = (float)i2;
    }
  }
}

// ---------------------------------------------------------------------------
extern "C" void kernel_launch(void* const* d_in, const int* in_sizes, int n_in,
                              void* d_out, int out_size, void* d_ws, size_t ws_size,
                              hipStream_t stream) {
  const float* pc    = (const float*)d_in[0];  // (4,16384,3)  f32
  const float* feats = (const float*)d_in[1];  // (4,16384,32) f32
  const int*   rnds  = (const int*)d_in[2];    // (1,16384)    i32
  float* out = (float*)d_out;

  (void)in_sizes; (void)n_in; (void)out_size; (void)d_ws; (void)ws_size;

  int gblocks = GATHER_THREADS / 256;  // 4544
  gather_kernel<<<gblocks, 256, 0, stream>>>(pc, feats, rnds, out);

  // 4 batches * (8192 queries / 128 per block) = 256 blocks
  knn_kernel<<<BATCH * (HALF / 128), 256, 0, stream>>>(pc, rnds, out);
}